// FM_MTL_28733331210610
// MI455X (gfx1250) — compile-verified
//
#include <hip/hip_runtime.h>
#include <hip/hip_bf16.h>
#include <math.h>

// ---------------------------------------------------------------------------
// DeepFM forward (FM + two sigmoid heads), MI455X / gfx1250, wave32.
//
// Roofline: ~109MB of gathered 256B emb2 rows dominates -> pure gather
// bandwidth bound (~5us @ 23.3TB/s). FLOPs (~0.1G) are noise.
//
// Kernel 1: dense 13->1 linear layer on matrix cores (V_WMMA_F32_16X16X4_F32,
//           exact f32 GEMV). Full tiles only; operand masking done with
//           mask-multiplies (not selects) so no load can be sunk under an
//           exec-mask branch: straight-line loads -> 4xWMMA -> store.
// Kernel 2: one wave per batch row; each 256B embedding row is fetched as
//           16 lanes x global_load_b128 (two fields per iteration), FM
//           sum-square trick + wave reduction + sigmoid heads.
// ---------------------------------------------------------------------------

#define N_DENSE 13
#define N_SPARSE 26
#define VOCAB 100000
#define EMB 64

typedef __attribute__((ext_vector_type(2))) float v2f;
typedef __attribute__((ext_vector_type(8))) float v8f;

// ---------------------------------------------------------------------------
// Kernel 1: lin_out[b] = dense[b,:] @ W_dense + b_dense via WMMA, full tiles.
//
// A layout (16x4 f32, ISA 7.12.2): lane l<16 -> M=l, {K=k0,k0+1}; lanes
// 16..31 -> same M, {K=k0+2,k0+3}. B (4x16) symmetric, N striped across
// lanes; W lives in column N=0 only, so D[m,0] = dot. C preloaded with bias.
// kk=0..2 cover K=0..11 (all in-bounds, unconditional); kk=3 covers K=12..15
// where only K=12 is real (half==0 .x slot). Lane masking is done by
// multiplying unconditionally-loaded finite values by {0,1} masks, so clang
// cannot sink any load under an exec-masked branch.
// D extraction: VGPR r of lane 0 -> (m=r, n=0); lane 16 -> (m=8+r, n=0).
// ---------------------------------------------------------------------------
__global__ void __launch_bounds__(32)
dense_gemv_wmma_kernel(const float* __restrict__ dense,
                       const float* __restrict__ W_dense,
                       const float* __restrict__ b_dense,
                       float* __restrict__ lin_out) {
    const int lane = threadIdx.x;          // 0..31 (wave32)
    const int l    = lane & 15;            // M row (A) / N col (B)
    const int half = lane >> 4;            // K sub-split within a WMMA step
    const int row  = blockIdx.x * 16 + l;  // always in-bounds (full tiles)
    const float* drow = dense + row * N_DENSE;

    const float mB = (l == 0)    ? 1.0f : 0.0f;  // B column n==0 mask
    const float mH = (half == 0) ? 1.0f : 0.0f;  // low-K-half mask (kk=3 pad)

    const float bd = b_dense[0];
    v8f c = {bd, bd, bd, bd, bd, bd, bd, bd};

    // kk = 0..2: K = kk*4 + half*2 + {0,1}, all < 13 -> unconditional loads
#pragma unroll
    for (int kk = 0; kk < 3; ++kk) {
        const int k0 = kk * 4 + half * 2;
        v2f a, b;
        a.x = drow[k0];
        a.y = drow[k0 + 1];
        b.x = W_dense[k0]     * mB;        // load all lanes, mask by multiply
        b.y = W_dense[k0 + 1] * mB;
        c = __builtin_amdgcn_wmma_f32_16x16x4_f32(
                false, a, false, b, (short)0, c, false, false);
    }

    // kk = 3: only K=12 is real; it sits in the .x slot of half==0 lanes
    {
        const float d12 = drow[12];        // unconditional, in-bounds
        const float w12 = W_dense[12];
        v2f a, b;
        a.x = d12 * mH;
        a.y = 0.0f;
        b.x = w12 * (mB * mH);
        b.y = 0.0f;
        c = __builtin_amdgcn_wmma_f32_16x16x4_f32(
                false, a, false, b, (short)0, c, false, false);
    }

    // lanes 0 and 16 hold column N=0 of D (8 rows each); 2x float4 stores
    if (l == 0) {
        const int mbase = blockIdx.x * 16 + half * 8;
        *(float4*)(lin_out + mbase)     = make_float4(c[0], c[1], c[2], c[3]);
        *(float4*)(lin_out + mbase + 4) = make_float4(c[4], c[5], c[6], c[7]);
    }
}

// Scalar fallback for a partial tail tile (batch % 16 != 0; unused at B=16384)
__global__ void __launch_bounds__(64)
dense_gemv_tail_kernel(const float* __restrict__ dense,
                       const float* __restrict__ W_dense,
                       const float* __restrict__ b_dense,
                       float* __restrict__ lin_out, int start, int batch) {
    const int row = start + blockIdx.x * blockDim.x + threadIdx.x;
    if (row >= batch) return;
    float acc = b_dense[0];
#pragma unroll
    for (int k = 0; k < N_DENSE; ++k) acc += dense[row * N_DENSE + k] * W_dense[k];
    lin_out[row] = acc;
}

// ---------------------------------------------------------------------------
// Kernel 2: one wave per batch row; gather-bound FM body.
//   lanes 0..25 own one field: its index + its emb1 scalar.
//   Each iteration fetches TWO 256B rows: lanes 0-15 -> field 2i (float4 per
//   lane), lanes 16-31 -> field 2i+1. 13 global_load_b128 per wave total.
//   FM: per-dim totals combined across halves via shfl_xor(16); squared term
//   weighted 0.25 (counted twice over 32 lanes), sum-of-squares term 0.5.
// ---------------------------------------------------------------------------
__global__ void __launch_bounds__(256)
fm_main_kernel(const int*   __restrict__ sparse,
               const float* __restrict__ emb1,
               const float* __restrict__ emb2,
               const float* __restrict__ lin_dense,   // ws from kernel 1
               const float* __restrict__ W_finish, const float* __restrict__ b_finish,
               const float* __restrict__ W_like,   const float* __restrict__ b_like,
               float* __restrict__ out, int batch) {
    const int lane = threadIdx.x & 31;
    const int b    = (blockIdx.x * blockDim.x + threadIdx.x) >> 5;
    if (b >= batch) return;                 // uniform per wave

    const int g = lane >> 4;                // half-wave id: field parity
    const int j = lane & 15;                // float4 slot within the 256B row

    // lanes 0..25 own one field: index + first-order emb1 scalar
    int   myidx = 0;
    float lin   = 0.0f;
    if (lane < N_SPARSE) {
        myidx = sparse[b * N_SPARSE + lane];
        lin   = emb1[(size_t)lane * VOCAB + (size_t)myidx];
    }

    float4 s = make_float4(0.f, 0.f, 0.f, 0.f);   // per-dim sums (4 dims/lane)
    float4 q = make_float4(0.f, 0.f, 0.f, 0.f);   // per-dim sums of squares
#pragma unroll
    for (int i = 0; i < N_SPARSE / 2; ++i) {
        const int f    = 2 * i + g;               // this half-wave's field
        const int idxf = __shfl(myidx, f, 32);
        const float4* row =
            (const float4*)(emb2 + ((size_t)f * VOCAB + (size_t)idxf) * EMB);
        const float4 v = row[j];                  // 16 lanes x 16B = 256B line
        s.x += v.x; s.y += v.y; s.z += v.z; s.w += v.w;
        q.x += v.x * v.x; q.y += v.y * v.y; q.z += v.z * v.z; q.w += v.w * v.w;
    }

    // combine even/odd field partial sums across half-waves (dims align)
    s.x += __shfl_xor(s.x, 16, 32);
    s.y += __shfl_xor(s.y, 16, 32);
    s.z += __shfl_xor(s.z, 16, 32);
    s.w += __shfl_xor(s.w, 16, 32);

    // totals duplicated across halves -> 0.25 weight nets to 0.5*sum over dims
    float part = lin
               + 0.25f * (s.x * s.x + s.y * s.y + s.z * s.z + s.w * s.w)
               - 0.5f  * (q.x + q.y + q.z + q.w);

#pragma unroll
    for (int m = 16; m >= 1; m >>= 1)
        part += __shfl_xor(part, m, 32);

    if (lane == 0) {
        const float logits = part + lin_dense[b];   // dense linear from WMMA
        const float zf = logits * W_finish[0] + b_finish[0];
        const float zl = logits * W_like[0]   + b_like[0];
        out[b]         = 1.0f / (1.0f + expf(-zf));
        out[batch + b] = 1.0f / (1.0f + expf(-zl));
    }
}

extern "C" void kernel_launch(void* const* d_in, const int* in_sizes, int n_in,
                              void* d_out, int out_size, void* d_ws, size_t ws_size,
                              hipStream_t stream) {
    const int*   sparse   = (const int*)  d_in[0];  // [B, 26] int32
    const float* dense    = (const float*)d_in[1];  // [B, 13] f32
    const float* emb1     = (const float*)d_in[2];  // [26, V, 1]
    const float* emb2     = (const float*)d_in[3];  // [26, V, 64]
    const float* W_dense  = (const float*)d_in[4];  // [13, 1]
    const float* b_dense  = (const float*)d_in[5];  // [1]
    const float* W_finish = (const float*)d_in[6];  // [1,1]
    const float* b_finish = (const float*)d_in[7];  // [1]
    const float* W_like   = (const float*)d_in[8];  // [1,1]
    const float* b_like   = (const float*)d_in[9];  // [1]

    const int batch = in_sizes[0] / N_SPARSE;       // 16384
    float* lin_dense = (float*)d_ws;                // batch floats (64 KB)
    float* out = (float*)d_out;                     // [finish(B) | like(B)]

    // 1) dense 13->1 linear layer on matrix cores (writes ws)
    const int fullTiles = batch / 16;
    if (fullTiles > 0)
        dense_gemv_wmma_kernel<<<fullTiles, 32, 0, stream>>>(dense, W_dense,
                                                             b_dense, lin_dense);
    const int rem = batch - fullTiles * 16;
    if (rem > 0)
        dense_gemv_tail_kernel<<<1, 64, 0, stream>>>(dense, W_dense, b_dense,
                                                     lin_dense, fullTiles * 16,
                                                     batch);

    // 2) gather-bound FM body: one wave per batch row, 8 waves per block
    const int threads = 256;
    const int blocks  = (batch * 32 + threads - 1) / threads;
    fm_main_kernel<<<blocks, threads, 0, stream>>>(sparse, emb1, emb2, lin_dense,
                                                   W_finish, b_finish,
                                                   W_like, b_like, out, batch);
}